// ContextEmbedding_31971736551608
// MI455X (gfx1250) — compile-verified
//
#include <hip/hip_runtime.h>
#include <hip/hip_bf16.h>

// ---------------------------------------------------------------------------
// ContextEmbedding for MI455X (gfx1250, wave32).
// Combined-branch matmul via V_WMMA_F32_16X16X4_F32 (full fp32 precision):
//   cw[20][256] = [ctx_w(16) ; cls_w(3) ; 0], A rows one-hot-masked per token.
// Memory-bound: 128MB out + 8MB in -> ~5.8us floor at 23.3 TB/s.
// ---------------------------------------------------------------------------

typedef float v2f __attribute__((ext_vector_type(2)));
typedef float v8f __attribute__((ext_vector_type(8)));

#define D_MODEL       256
#define NUM_CTX       16
#define SPECIAL_OFF   68
#define NUM_SPECIAL   8
#define TOK_CLS       68   // SPECIAL_OFF + 0
#define TOK_CTX       69   // SPECIAL_OFF + 1
#define LN_EPS        1e-5f
#define HSTRIDE       260  // 256+4 pad: half-wave C rows -> distinct banks; 16B aligned
#define CWSTRIDE      264  // 256+8 pad: K and K+2 rows -> distinct banks
#define WAVES_PER_BLK 2
#define TOK_PER_WAVE  16

__global__ __launch_bounds__(WAVES_PER_BLK * 32)
void context_embedding_kernel(const int*   __restrict__ token_ids,
                              const float* __restrict__ ctx_feat,   // [B*S, 16]
                              const float* __restrict__ special_tab,// [8, 256]
                              const float* __restrict__ cls_w,      // [3, 256]
                              const float* __restrict__ cls_b,
                              const float* __restrict__ cls_g,
                              const float* __restrict__ cls_beta,
                              const float* __restrict__ ctx_w,      // [16, 256]
                              const float* __restrict__ ctx_b,
                              const float* __restrict__ ctx_g,
                              const float* __restrict__ ctx_beta,
                              float*       __restrict__ out)        // [B*S, 256]
{
    __shared__ float s_h[WAVES_PER_BLK][TOK_PER_WAVE][HSTRIDE]; // pre-LN activations
    __shared__ float s_cw[20][CWSTRIDE];  // combined weights: ctx_w(16) ; cls_w(3) ; 0
    __shared__ float s_par[6][D_MODEL];   // 0:cls_b 1:cls_g 2:cls_beta 3:ctx_b 4:ctx_g 5:ctx_beta
    __shared__ int   s_ids[WAVES_PER_BLK][TOK_PER_WAVE];
    __shared__ float s_mu[WAVES_PER_BLK][TOK_PER_WAVE];
    __shared__ float s_rstd[WAVES_PER_BLK][TOK_PER_WAVE];

    const int tid  = threadIdx.x;
    const int wave = tid >> 5;
    const int lane = tid & 31;
    const int ln   = lane & 15;     // N column within tile / token row for A
    const int hi   = lane >> 4;     // half-wave select (K split for A/B frags)

    // ---- stage combined weights + LN params into LDS (once per block) ----
    {
        for (int i = tid; i < 20 * D_MODEL; i += WAVES_PER_BLK * 32) {
            const int k = i >> 8, d = i & 255;
            float w;
            if (k < 16)      w = ctx_w[k * D_MODEL + d];
            else if (k < 19) w = cls_w[(k - 16) * D_MODEL + d];
            else             w = 0.0f;
            s_cw[k][d] = w;
        }
        const float* parr[6] = { cls_b, cls_g, cls_beta, ctx_b, ctx_g, ctx_beta };
        #pragma unroll
        for (int a = 0; a < 6; ++a)
            for (int i = tid; i < D_MODEL; i += WAVES_PER_BLK * 32)
                s_par[a][i] = parr[a][i];
    }

    const int group = blockIdx.x * WAVES_PER_BLK + wave;     // 16-token group
    const int base  = group * TOK_PER_WAVE;                  // flat token index

    if (lane < TOK_PER_WAVE) s_ids[wave][lane] = token_ids[base + lane];
    __syncthreads();

    // ---- build per-lane A fragments (32-bit A 16x4 layout, 5 K-steps) ----
    // lane (ln,hi) owns token M=ln; VGPR0/1 hold K = 4s+2*hi, 4s+2*hi+1
    const int   my_tok = s_ids[wave][ln];
    const bool  aCls   = (my_tok == TOK_CLS);
    const bool  aCtx   = (my_tok == TOK_CTX);
    const float* xf = ctx_feat + (size_t)(base + ln) * NUM_CTX;
    float4 x0 = ((const float4*)xf)[0];
    float4 x1 = ((const float4*)xf)[1];
    float4 x2 = ((const float4*)xf)[2];
    float4 x3 = ((const float4*)xf)[3];

    float xa[20];
    xa[0]=x0.x; xa[1]=x0.y; xa[2]=x0.z; xa[3]=x0.w;
    xa[4]=x1.x; xa[5]=x1.y; xa[6]=x1.z; xa[7]=x1.w;
    xa[8]=x2.x; xa[9]=x2.y; xa[10]=x2.z; xa[11]=x2.w;
    xa[12]=x3.x; xa[13]=x3.y; xa[14]=x3.z; xa[15]=x3.w;
    #pragma unroll
    for (int k = 0; k < 16; ++k) xa[k] = aCtx ? xa[k] : 0.0f;
    xa[16] = aCls ? x0.x : 0.0f;
    xa[17] = aCls ? x0.y : 0.0f;
    xa[18] = aCls ? x0.z : 0.0f;
    xa[19] = 0.0f;

    v2f afr[5];
    #pragma unroll
    for (int s = 0; s < 5; ++s) {
        afr[s].x = hi ? xa[4 * s + 2] : xa[4 * s + 0];
        afr[s].y = hi ? xa[4 * s + 3] : xa[4 * s + 1];
    }

    // ---- 16 column tiles, 5 WMMA K-steps each (K=20 combined) ----
    #pragma unroll 4
    for (int t = 0; t < 16; ++t) {
        v8f acc = {0.f, 0.f, 0.f, 0.f, 0.f, 0.f, 0.f, 0.f};
        const int d = t * 16 + ln;
        const float* cwp = &s_cw[2 * hi][d];   // per-lane K-base (hi split), col d
        #pragma unroll
        for (int s = 0; s < 5; ++s) {
            // 32-bit B 4x16 layout: VGPR v, lanes 0-15 -> K=4s+v, lanes 16-31 -> K=4s+v+2
            v2f bfr;
            bfr.x = cwp[(4 * s + 0) * CWSTRIDE];
            bfr.y = cwp[(4 * s + 1) * CWSTRIDE];
            acc = __builtin_amdgcn_wmma_f32_16x16x4_f32(
                /*neg_a=*/false, afr[s], /*neg_b=*/false, bfr,
                /*c_mod=*/(short)0, acc, /*reuse_a=*/false, /*reuse_b=*/false);
        }
        // C layout: VGPR v -> row M = v + 8*hi, col N = ln
        #pragma unroll
        for (int v = 0; v < 8; ++v)
            s_h[wave][v + 8 * hi][d] = acc[v];
    }

    // ---- LayerNorm statistics: 2 lanes per token, 128 dims each (float4) ----
    {
        const int  t2    = lane >> 1;
        const int  half  = lane & 1;
        const int  tk    = s_ids[wave][t2];
        const int  po    = (tk == TOK_CLS) ? 0 : 3;   // bias select (don't-care if unmasked)
        const float4* hrow = (const float4*)&s_h[wave][t2][half * 128];
        const float4* brow = (const float4*)&s_par[po][half * 128];
        float sum = 0.f, sq = 0.f;
        #pragma unroll 8
        for (int j = 0; j < 32; ++j) {
            const float4 h4 = hrow[j];
            const float4 b4 = brow[j];
            const float e0 = h4.x + b4.x, e1 = h4.y + b4.y;
            const float e2 = h4.z + b4.z, e3 = h4.w + b4.w;
            sum += (e0 + e1) + (e2 + e3);
            sq  += (e0 * e0 + e1 * e1) + (e2 * e2 + e3 * e3);
        }
        sum += __shfl_xor(sum, 1, 32);
        sq  += __shfl_xor(sq, 1, 32);
        const float mu  = sum * (1.0f / D_MODEL);
        float var = sq * (1.0f / D_MODEL) - mu * mu;
        var = fmaxf(var, 0.0f);
        const float rstd = rsqrtf(var + LN_EPS);
        if (half == 0) { s_mu[wave][t2] = mu; s_rstd[wave][t2] = rstd; }
    }
    __syncthreads();

    // ---- normalize + ReLU + special-table add, coalesced float4 stores ----
    #pragma unroll 2
    for (int t = 0; t < 16; ++t) {
        const int  tk     = __builtin_amdgcn_readfirstlane(s_ids[wave][t]); // wave-uniform
        const bool isSpec = (tk >= SPECIAL_OFF) && (tk < SPECIAL_OFF + NUM_SPECIAL);
        const int  sid    = isSpec ? (tk - SPECIAL_OFF) : 0;
        const bool isCls  = (tk == TOK_CLS);
        const bool isBr   = isCls || (tk == TOK_CTX);
        const int  po     = isCls ? 0 : 3;
        const float mu    = s_mu[wave][t];
        const float rstd  = s_rstd[wave][t];
        float* op = out + (size_t)(base + t) * D_MODEL;
        const float* sp = special_tab + sid * D_MODEL;
        #pragma unroll
        for (int rep = 0; rep < 2; ++rep) {
            const int d0 = lane * 4 + rep * 128;
            float4 o = {0.f, 0.f, 0.f, 0.f};
            if (isBr) {
                const float4 h4 = *(const float4*)&s_h[wave][t][d0];
                const float4 b4 = *(const float4*)&s_par[po][d0];
                const float4 g4 = *(const float4*)&s_par[po + 1][d0];
                const float4 e4 = *(const float4*)&s_par[po + 2][d0];
                o.x = fmaxf((h4.x + b4.x - mu) * rstd * g4.x + e4.x, 0.f);
                o.y = fmaxf((h4.y + b4.y - mu) * rstd * g4.y + e4.y, 0.f);
                o.z = fmaxf((h4.z + b4.z - mu) * rstd * g4.z + e4.z, 0.f);
                o.w = fmaxf((h4.w + b4.w - mu) * rstd * g4.w + e4.w, 0.f);
            }
            if (isSpec) {
                const float4 s4 = *(const float4*)(sp + d0);   // L2-resident, coalesced
                o.x += s4.x; o.y += s4.y; o.z += s4.z; o.w += s4.w;
            }
            *(float4*)(op + d0) = o;
        }
    }
}

extern "C" void kernel_launch(void* const* d_in, const int* in_sizes, int n_in,
                              void* d_out, int out_size, void* d_ws, size_t ws_size,
                              hipStream_t stream) {
    const int*   token_ids   = (const int*)  d_in[0];
    const float* ctx_feat    = (const float*)d_in[1];
    const float* special_tab = (const float*)d_in[2];
    const float* cls_w       = (const float*)d_in[3];
    const float* cls_b       = (const float*)d_in[4];
    const float* cls_g       = (const float*)d_in[5];
    const float* cls_beta    = (const float*)d_in[6];
    const float* ctx_w       = (const float*)d_in[7];
    const float* ctx_b       = (const float*)d_in[8];
    const float* ctx_g       = (const float*)d_in[9];
    const float* ctx_beta    = (const float*)d_in[10];
    float* out = (float*)d_out;

    const int n_tokens = in_sizes[0];                       // B*S = 131072
    const int groups   = n_tokens / TOK_PER_WAVE;           // 8192
    const int blocks   = groups / WAVES_PER_BLK;            // 4096

    context_embedding_kernel<<<blocks, WAVES_PER_BLK * 32, 0, stream>>>(
        token_ids, ctx_feat, special_tab,
        cls_w, cls_b, cls_g, cls_beta,
        ctx_w, ctx_b, ctx_g, ctx_beta, out);
}